// WindowSelfAttention_37220186587446
// MI455X (gfx1250) — compile-verified
//
#include <hip/hip_runtime.h>

// CDNA5 / gfx1250, wave32. Fused windowed self-attention:
//   Phase 0: one-shot reformat of Wq/Wk/Wv into WMMA-fragment order (b128 loads)
//   Phase 1: Q,K,V = X @ W{q,k,v} + b  via V_WMMA_F32_16X16X4_F32 (exact fp32)
//   Phase 2: 3x3 window attention, wave-per-pixel, shfl_xor reductions.
//
// Round-3 change: A fragments are re-loaded from X (L1-resident, 8KB/wave)
// inside the k-loop instead of being held in a 64-VGPR array -- the previous
// version spilled that array to scratch (scratch_load_b128 in the hot loop).

typedef float v2f __attribute__((ext_vector_type(2)));
typedef float v8f __attribute__((ext_vector_type(8)));

#define DD 128          // feature dim
#define NT 8            // 128 / 16 column tiles
// Fragment-ordered W: per matrix, [kkp(16)][t(8)][lane(32)] float4
//   float4 = { W[k0+0][n], W[k0+1][n], W[k1+0][n], W[k1+1][n] }
//   k0 = 8*kkp + 2*hi, k1 = k0 + 4, n = t*16 + (lane&15), hi = lane>>4
#define WFRAG_F4_PER_MAT 4096   // 16*8*32

// ---------------------------------------------------------------------------
// Phase 0: gather W into fragment order so the GEMM loads b128 per 2 k-steps.
// ---------------------------------------------------------------------------
__global__ __launch_bounds__(256)
void reformat_w(const float* __restrict__ Wq, const float* __restrict__ Wk,
                const float* __restrict__ Wv, float4* __restrict__ Wf)
{
    const int idx = (int)(blockIdx.x * blockDim.x + threadIdx.x); // 0..12287
    const int m    = idx >> 12;          // matrix 0..2
    const int r    = idx & 4095;
    const int kkp  = r >> 8;             // 0..15
    const int t    = (r >> 5) & 7;       // 0..7
    const int lane = r & 31;
    const int hi = lane >> 4, l = lane & 15;
    const int n  = t * 16 + l;
    const float* W = (m == 0) ? Wq : ((m == 1) ? Wk : Wv);
    const int k0 = 8 * kkp + 2 * hi;
    const int k1 = k0 + 4;
    float4 v;
    v.x = W[(size_t)(k0 + 0) * DD + n];
    v.y = W[(size_t)(k0 + 1) * DD + n];
    v.z = W[(size_t)(k1 + 0) * DD + n];
    v.w = W[(size_t)(k1 + 1) * DD + n];
    Wf[idx] = v;
}

// ---------------------------------------------------------------------------
// QKV GEMM: one wave computes a 16-row M-tile of Q, K and V (full 128 cols).
// A fragments re-loaded per matrix from the (L1-hot) X tile; B fragments
// come from the fragment-ordered Wf via b128 loads.
// ---------------------------------------------------------------------------
__global__ __launch_bounds__(256)
void qkv_gemm_wmma(const float* __restrict__ X,
                   const float4* __restrict__ Wf,
                   const float* __restrict__ bq, const float* __restrict__ bk,
                   const float* __restrict__ bv,
                   float* __restrict__ Q, float* __restrict__ K, float* __restrict__ V,
                   int Mrows)
{
    const int wave = (int)((blockIdx.x * blockDim.x + threadIdx.x) >> 5);
    const int lane = (int)(threadIdx.x & 31);
    const int m0   = wave * 16;
    if (m0 >= Mrows) return;                 // wave-uniform; grid is exact anyway

    const int hi = lane >> 4;                // 0: lanes 0-15, 1: lanes 16-31
    const int l  = lane & 15;

    // A fragment base: lane<16 -> row M=l, K = 4*kk+{0,1}; lane>=16 -> K = 4*kk+{2,3}
    const float* xrow = X + (size_t)(m0 + l) * DD + 2 * hi;

    const float* bm[3] = { bq, bk, bv };
    float*       Om[3] = { Q,  K,  V  };

#pragma unroll
    for (int m = 0; m < 3; ++m) {
        const float4* Wp = Wf + (size_t)m * WFRAG_F4_PER_MAT;
        const float*  bp = bm[m];
        float*        Op = Om[m];

        v8f c[NT];
#pragma unroll
        for (int t = 0; t < NT; ++t) {
            const float bias = bp[t * 16 + l];   // C col = t*16 + (lane&15)
#pragma unroll
            for (int r = 0; r < 8; ++r) c[t][r] = bias;
        }

#pragma unroll 2
        for (int kkp = 0; kkp < 16; ++kkp) {
            // A fragments for k-steps 2*kkp and 2*kkp+1 (two aligned b64 loads)
            const v2f a0 = *(const v2f*)(xrow + 8 * kkp);
            const v2f a1 = *(const v2f*)(xrow + 8 * kkp + 4);
            // one b128 per tile covers both k-steps
            float4 bf[NT];
#pragma unroll
            for (int t = 0; t < NT; ++t)
                bf[t] = Wp[(size_t)(kkp * NT + t) * 32 + lane];
#pragma unroll
            for (int t = 0; t < NT; ++t) {
                v2f b0; b0.x = bf[t].x; b0.y = bf[t].y;
                c[t] = __builtin_amdgcn_wmma_f32_16x16x4_f32(
                           false, a0, false, b0, (short)0, c[t], false, false);
            }
#pragma unroll
            for (int t = 0; t < NT; ++t) {
                v2f b1; b1.x = bf[t].z; b1.y = bf[t].w;
                c[t] = __builtin_amdgcn_wmma_f32_16x16x4_f32(
                           false, a1, false, b1, (short)0, c[t], false, false);
            }
        }

        // C/D layout: VGPR r -> row m0 + 8*hi + r, col t*16 + l
#pragma unroll
        for (int t = 0; t < NT; ++t) {
#pragma unroll
            for (int r = 0; r < 8; ++r) {
                Op[(size_t)(m0 + 8 * hi + r) * DD + t * 16 + l] = c[t][r];
            }
        }
    }
}

// ---------------------------------------------------------------------------
// 3x3 window attention: one wave per pixel; each lane owns 4 channels.
// Reference pads K,V with zeros: OOB neighbor -> score 0 (participates in
// softmax) and zero V contribution.
// ---------------------------------------------------------------------------
__global__ __launch_bounds__(256)
void win_attn(const float* __restrict__ Q, const float* __restrict__ K,
              const float* __restrict__ V, float* __restrict__ out,
              int B, int H, int W)
{
    const int wave = (int)((blockIdx.x * blockDim.x + threadIdx.x) >> 5);
    const int lane = (int)(threadIdx.x & 31);
    const int npix = B * H * W;
    if (wave >= npix) return;

    const int b   = wave / (H * W);
    const int rem = wave % (H * W);
    const int y   = rem / W;
    const int x   = rem % W;
    const float scale = 0.08838834764831845f;   // 128^-0.5

    const float4 q = *(const float4*)(Q + (size_t)wave * DD + lane * 4);

    float  sc[9];
    float4 vf[9];
#pragma unroll
    for (int dy = -1; dy <= 1; ++dy) {
#pragma unroll
        for (int dx = -1; dx <= 1; ++dx) {
            const int  i  = (dy + 1) * 3 + (dx + 1);
            const int  yy = y + dy, xx = x + dx;
            const bool ok = (yy >= 0) & (yy < H) & (xx >= 0) & (xx < W);
            float  s  = 0.0f;
            float4 v4 = make_float4(0.f, 0.f, 0.f, 0.f);
            if (ok) {
                const size_t off = ((size_t)b * H * W + (size_t)yy * W + xx) * DD + lane * 4;
                const float4 k4 = *(const float4*)(K + off);
                v4 = *(const float4*)(V + off);
                s = q.x * k4.x + q.y * k4.y + q.z * k4.z + q.w * k4.w;
            }
            // wave32 butterfly reduce -> all lanes hold the full dot product
#pragma unroll
            for (int m = 16; m >= 1; m >>= 1) s += __shfl_xor(s, m, 32);
            sc[i] = s * scale;                 // OOB: exactly 0, matches zero-pad
            vf[i] = v4;
        }
    }

    float mx = sc[0];
#pragma unroll
    for (int i = 1; i < 9; ++i) mx = fmaxf(mx, sc[i]);
    float p[9], sum = 0.0f;
#pragma unroll
    for (int i = 0; i < 9; ++i) { p[i] = expf(sc[i] - mx); sum += p[i]; }
    const float inv = 1.0f / sum;

    float4 o = make_float4(0.f, 0.f, 0.f, 0.f);
#pragma unroll
    for (int i = 0; i < 9; ++i) {
        const float w = p[i] * inv;
        o.x += w * vf[i].x; o.y += w * vf[i].y;
        o.z += w * vf[i].z; o.w += w * vf[i].w;
    }
    *(float4*)(out + (size_t)wave * DD + lane * 4) = o;
}

// ---------------------------------------------------------------------------
extern "C" void kernel_launch(void* const* d_in, const int* in_sizes, int n_in,
                              void* d_out, int out_size, void* d_ws, size_t ws_size,
                              hipStream_t stream)
{
    const float* x  = (const float*)d_in[0];
    const float* Wq = (const float*)d_in[1];
    const float* bq = (const float*)d_in[2];
    const float* Wk = (const float*)d_in[3];
    const float* bk = (const float*)d_in[4];
    const float* Wv = (const float*)d_in[5];
    const float* bv = (const float*)d_in[6];

    const int H = 128, W = 128;
    const int Mrows = in_sizes[0] / DD;       // B * N = 131072
    const int B = Mrows / (H * W);

    float*  Q   = (float*)d_ws;               // 3 * Mrows * 128 * 4 = 192 MB
    float*  Kp  = Q  + (size_t)Mrows * DD;
    float*  Vp  = Kp + (size_t)Mrows * DD;
    float4* Wf  = (float4*)(Vp + (size_t)Mrows * DD);   // + 192 KB fragment-ordered W

    // Phase 0: 3 * 4096 float4 elements, 12288 threads
    reformat_w<<<48, 256, 0, stream>>>(Wq, Wk, Wv, Wf);

    const int gemmWaves = Mrows / 16;         // 8192 (exact)
    qkv_gemm_wmma<<<(gemmWaves + 7) / 8, 256, 0, stream>>>(
        x, Wf, bq, bk, bv, Q, Kp, Vp, Mrows);

    const int pixWaves = Mrows;               // 131072 (exact)
    win_attn<<<(pixWaves + 7) / 8, 256, 0, stream>>>(
        Q, Kp, Vp, (float*)d_out, B, H, W);
}